// MultiHeadedAttention_25237227831700
// MI455X (gfx1250) — compile-verified
//
#include <hip/hip_runtime.h>
#include <hip/hip_bf16.h>

typedef float v2f __attribute__((ext_vector_type(2)));
typedef float v8f __attribute__((ext_vector_type(8)));

#define MDIM 16384   // B*S
#define NDIM 1024    // D
#define KDIM 1024    // D
#define BATCH 4
#define SEQ 4096
#define HEADS 16
#define DK 64

#define LDS_STRIDE 68   // 64 + 4: lane-halves (rows m and m+8) land on disjoint banks

// ---------------------------------------------------------------------------
// Core WMMA fp32 GEMM tile:  Y[m,n] = sum_k X[m,k] * W[n,k] + bias[n]
// Block = 128 threads (4 waves); wave computes a 64x64 tile with a 4x4 grid
// of v_wmma_f32_16x16x4_f32. Epilogue stages the tile in LDS and emits
// coalesced float4 stores.
// ---------------------------------------------------------------------------
__device__ __forceinline__ void gemm64_tile(
    const float* __restrict__ X, const float* __restrict__ W,
    const float* __restrict__ bias, float* __restrict__ Y,
    float* __restrict__ lds /* per-wave 64*LDS_STRIDE floats */,
    int bx, int by)
{
    const int lane = threadIdx.x & 31;
    const int wave = threadIdx.x >> 5;      // 0..3
    const int r    = lane & 15;
    const int h2   = lane >> 4;             // lane-half

    const int tileM = bx * 128 + (wave >> 1) * 64;
    const int tileN = by * 128 + (wave & 1) * 64;

    v8f acc[4][4];
#pragma unroll
    for (int i = 0; i < 4; ++i)
#pragma unroll
        for (int j = 0; j < 4; ++j)
            acc[i][j] = (v8f)(0.0f);

    // A frag (16x4 f32): lane holds A[r][2*h2 + 0/1] -> contiguous float2 of X
    // B frag (4x16 f32): lane holds B[2*h2 + 0/1][r] = W[col r][2*h2 + 0/1]
    const float* xrow[4];
    const float* wrow[4];
#pragma unroll
    for (int i = 0; i < 4; ++i)
        xrow[i] = X + (size_t)(tileM + i * 16 + r) * KDIM + 2 * h2;
#pragma unroll
    for (int j = 0; j < 4; ++j)
        wrow[j] = W + (size_t)(tileN + j * 16 + r) * KDIM + 2 * h2;

#pragma unroll 2
    for (int k = 0; k < KDIM; k += 4) {
        v2f a[4], b[4];
#pragma unroll
        for (int i = 0; i < 4; ++i)
            a[i] = *(const v2f*)(xrow[i] + k);
#pragma unroll
        for (int j = 0; j < 4; ++j)
            b[j] = *(const v2f*)(wrow[j] + k);
#pragma unroll
        for (int i = 0; i < 4; ++i)
#pragma unroll
            for (int j = 0; j < 4; ++j)
                acc[i][j] = __builtin_amdgcn_wmma_f32_16x16x4_f32(
                    false, a[i], false, b[j], (short)0, acc[i][j], false, false);
    }

    // --- Epilogue: acc (lane holds col n=r, rows v+8*h2) -> LDS tile ---
#pragma unroll
    for (int i = 0; i < 4; ++i)
#pragma unroll
        for (int j = 0; j < 4; ++j) {
            const int nn = j * 16 + r;
#pragma unroll
            for (int v = 0; v < 8; ++v) {
                const int mm = i * 16 + v + 8 * h2;
                lds[mm * LDS_STRIDE + nn] = acc[i][j][v];
            }
        }

    // --- Read back row-major + bias, coalesced float4 global stores ---
    const int coln = (lane & 15) * 4;                 // 0..60
    const float4 bv = *(const float4*)(bias + tileN + coln);
#pragma unroll
    for (int rr = 0; rr < 32; ++rr) {
        const int row = rr * 2 + h2;                  // 0..63
        float4 t = *(const float4*)(lds + row * LDS_STRIDE + coln);
        t.x += bv.x; t.y += bv.y; t.z += bv.z; t.w += bv.w;
        *(float4*)(Y + (size_t)(tileM + row) * NDIM + tileN + coln) = t;
    }
}

// One launch computes all three QKV projections (blockIdx.z selects).
__global__ __launch_bounds__(128) void gemm_qkv_wmma(
    const float* __restrict__ Xq, const float* __restrict__ Xk, const float* __restrict__ Xv,
    const float* __restrict__ Wq, const float* __restrict__ Wk, const float* __restrict__ Wv,
    const float* __restrict__ bq, const float* __restrict__ bk, const float* __restrict__ bv,
    float* __restrict__ Yq, float* __restrict__ Yk, float* __restrict__ Yv)
{
    __shared__ float lds[4][64 * LDS_STRIDE];
    const int wave = threadIdx.x >> 5;
    const float* X; const float* W; const float* B; float* Y;
    if (blockIdx.z == 0)      { X = Xq; W = Wq; B = bq; Y = Yq; }
    else if (blockIdx.z == 1) { X = Xk; W = Wk; B = bk; Y = Yk; }
    else                      { X = Xv; W = Wv; B = bv; Y = Yv; }
    gemm64_tile(X, W, B, Y, lds[wave], blockIdx.x, blockIdx.y);
}

__global__ __launch_bounds__(128) void gemm_single_wmma(
    const float* __restrict__ X, const float* __restrict__ W,
    const float* __restrict__ bias, float* __restrict__ Y)
{
    __shared__ float lds[4][64 * LDS_STRIDE];
    gemm64_tile(X, W, bias, Y, lds[threadIdx.x >> 5], blockIdx.x, blockIdx.y);
}

// ---------------------------------------------------------------------------
// k_sum two-phase deterministic reduction over S.
// Phase A: grid (16, 64): partial[split][b*D+d] = sum of 64 s-values.
// Phase B: grid 16: ksum[b*D+d] = sum over 64 partials (fixed order).
// ---------------------------------------------------------------------------
__global__ __launch_bounds__(256) void ksum_partial_kernel(
    const float* __restrict__ kbuf, float* __restrict__ partial)
{
    const int idx = blockIdx.x * 256 + threadIdx.x;   // 0..4095 over B*D
    const int b = idx / NDIM;
    const int d = idx % NDIM;
    const int s0 = blockIdx.y * 64;
    const float* col = kbuf + ((size_t)b * SEQ + s0) * NDIM + d;
    float acc = 0.0f;
#pragma unroll 4
    for (int s = 0; s < 64; ++s)
        acc += col[(size_t)s * NDIM];
    partial[(size_t)blockIdx.y * (BATCH * NDIM) + idx] = acc;
}

__global__ __launch_bounds__(256) void ksum_combine_kernel(
    const float* __restrict__ partial, float* __restrict__ ksum)
{
    const int idx = blockIdx.x * 256 + threadIdx.x;
    float acc = 0.0f;
#pragma unroll
    for (int k = 0; k < 64; ++k)
        acc += partial[(size_t)k * (BATCH * NDIM) + idx];
    ksum[idx] = acc;
}

// ---------------------------------------------------------------------------
// Per (b,h): scores[s] = q[b,h,s,:] . ksum[b,h,:]; softmax over s -> p_out.
// ---------------------------------------------------------------------------
__global__ __launch_bounds__(256) void score_softmax_kernel(
    const float* __restrict__ qbuf, const float* __restrict__ ksum,
    float* __restrict__ p_out)
{
    __shared__ float sm_scores[SEQ];
    __shared__ float sm_ks[DK];
    __shared__ float red[256];

    const int bh = blockIdx.x;         // 0..63
    const int b  = bh >> 4;
    const int h  = bh & 15;
    const int tid = threadIdx.x;

    if (tid < DK)
        sm_ks[tid] = ksum[b * NDIM + h * DK + tid];
    __syncthreads();

    const float4* ks4 = (const float4*)sm_ks;

    float lmax = -3.402823466e+38f;
#pragma unroll
    for (int i = 0; i < 16; ++i) {
        const int s = tid + i * 256;
        const float4* q4 = (const float4*)(qbuf + (size_t)(b * SEQ + s) * NDIM + h * DK);
        float dot = 0.0f;
#pragma unroll
        for (int c = 0; c < DK / 4; ++c) {
            const float4 qv = q4[c];
            const float4 kv = ks4[c];
            dot = fmaf(qv.x, kv.x, dot);
            dot = fmaf(qv.y, kv.y, dot);
            dot = fmaf(qv.z, kv.z, dot);
            dot = fmaf(qv.w, kv.w, dot);
        }
        sm_scores[s] = dot;
        lmax = fmaxf(lmax, dot);
    }

    red[tid] = lmax;
    __syncthreads();
    for (int off = 128; off > 0; off >>= 1) {
        if (tid < off) red[tid] = fmaxf(red[tid], red[tid + off]);
        __syncthreads();
    }
    const float gmax = red[0];
    __syncthreads();

    float lsum = 0.0f;
#pragma unroll
    for (int i = 0; i < 16; ++i) {
        const int s = tid + i * 256;
        const float e = __expf(sm_scores[s] - gmax);
        sm_scores[s] = e;
        lsum += e;
    }
    red[tid] = lsum;
    __syncthreads();
    for (int off = 128; off > 0; off >>= 1) {
        if (tid < off) red[tid] += red[tid + off];
        __syncthreads();
    }
    const float inv = 1.0f / red[0];

#pragma unroll
    for (int i = 0; i < 16; ++i) {
        const int s = tid + i * 256;
        p_out[(size_t)bh * SEQ + s] = sm_scores[s] * inv;
    }
}

// ---------------------------------------------------------------------------
// x[m, d] = p[b, d>>6, s] * v[m, d], float4-vectorized.
// ---------------------------------------------------------------------------
__global__ __launch_bounds__(256) void scale_v_kernel(
    const float* __restrict__ vbuf, const float* __restrict__ p,
    float* __restrict__ xbuf)
{
    const size_t idx4 = (size_t)blockIdx.x * 256 + threadIdx.x;
    const size_t base = idx4 * 4;
    const int m = (int)(base / NDIM);
    const int d = (int)(base % NDIM);
    const int b = m / SEQ;
    const int s = m % SEQ;
    const int head = d >> 6;
    const float pf = p[((size_t)(b * HEADS + head)) * SEQ + s];
    const float4 vv = ((const float4*)vbuf)[idx4];
    float4 xx;
    xx.x = pf * vv.x; xx.y = pf * vv.y; xx.z = pf * vv.z; xx.w = pf * vv.w;
    ((float4*)xbuf)[idx4] = xx;
}

// ---------------------------------------------------------------------------
extern "C" void kernel_launch(void* const* d_in, const int* in_sizes, int n_in,
                              void* d_out, int out_size, void* d_ws, size_t ws_size,
                              hipStream_t stream) {
    const float* query = (const float*)d_in[0];
    const float* key   = (const float*)d_in[1];
    const float* value = (const float*)d_in[2];
    const float* Wq = (const float*)d_in[3];
    const float* bq = (const float*)d_in[4];
    const float* Wk = (const float*)d_in[5];
    const float* bk = (const float*)d_in[6];
    const float* Wv = (const float*)d_in[7];
    const float* bv = (const float*)d_in[8];
    const float* Wo = (const float*)d_in[9];
    const float* bo = (const float*)d_in[10];

    float* out   = (float*)d_out;                       // [B,S,D]
    float* p_out = out + (size_t)MDIM * NDIM;           // [B,H,S,1]

    float* qbuf = (float*)d_ws;                         // 64 MB
    float* kbuf = qbuf + (size_t)MDIM * NDIM;           // 64 MB (reused as xbuf)
    float* vbuf = kbuf + (size_t)MDIM * NDIM;           // 64 MB
    float* ksum = vbuf + (size_t)MDIM * NDIM;           // 16 KB
    float* part = ksum + (size_t)BATCH * NDIM;          // 1 MB partials

    dim3 gblock(128, 1, 1);
    dim3 gqkv(MDIM / 128, NDIM / 128, 3);               // (128, 8, 3)
    dim3 gone(MDIM / 128, NDIM / 128, 1);

    gemm_qkv_wmma<<<gqkv, gblock, 0, stream>>>(
        query, key, value, Wq, Wk, Wv, bq, bk, bv, qbuf, kbuf, vbuf);

    ksum_partial_kernel<<<dim3((BATCH * NDIM) / 256, SEQ / 64, 1), 256, 0, stream>>>(kbuf, part);
    ksum_combine_kernel<<<(BATCH * NDIM) / 256, 256, 0, stream>>>(part, ksum);

    score_softmax_kernel<<<BATCH * HEADS, 256, 0, stream>>>(qbuf, ksum, p_out);

    const int scale_blocks = (MDIM * NDIM / 4) / 256;   // 16384
    scale_v_kernel<<<scale_blocks, 256, 0, stream>>>(vbuf, p_out, kbuf);

    gemm_single_wmma<<<gone, gblock, 0, stream>>>(kbuf, Wo, bo, out);
}